// DeepPrimalDual_88244398063723
// MI455X (gfx1250) — compile-verified
//
#include <hip/hip_runtime.h>
#include <hip/hip_bf16.h>
#include <math.h>

#define NITER_FIXED 5

typedef float v2f __attribute__((ext_vector_type(2)));
typedef float v8f __attribute__((ext_vector_type(8)));

__device__ __forceinline__ float fatomic_add(float* p, float v) {
  return __hip_atomic_fetch_add(p, v, __ATOMIC_RELAXED, __HIP_MEMORY_SCOPE_AGENT);
}

__global__ void zero_f32(float* __restrict__ p, long count) {
  long i = (long)blockIdx.x * blockDim.x + threadIdx.x;
  if (i < count) p[i] = 0.f;
}

// state5[i*5] = value, state5[i*5+1..4] = hidden (zeros at init)
__global__ void init_state5(float* __restrict__ s5, const float* __restrict__ v0, int count) {
  int i = blockIdx.x * blockDim.x + threadIdx.x;
  if (i < count) {
    long b = (long)i * 5;
    s5[b + 0] = v0[i];
    s5[b + 1] = 0.f; s5[b + 2] = 0.f; s5[b + 3] = 0.f; s5[b + 4] = 0.f;
  }
}

// dst5[scatter[e]*5+k] += vals[e] * src5[gather[e]*5+k]  (k=0..4)
// A@[x|x_h]: gather=cols, scatter=rows.  A.T@[y|y_h]: gather=rows, scatter=cols.
__global__ void spmv5(const int* __restrict__ gather_idx, const int* __restrict__ scatter_idx,
                      const float* __restrict__ vals, const float* __restrict__ src5,
                      float* __restrict__ dst5, int nnz) {
  int e = blockIdx.x * blockDim.x + threadIdx.x;
  if (e >= nnz) return;
  float v = vals[e];
  long sb = (long)gather_idx[e] * 5;
  long db = (long)scatter_idx[e] * 5;
#pragma unroll
  for (int k = 0; k < 5; ++k) fatomic_add(&dst5[db + k], v * src5[sb + k]);
}

// Fused tower update. Each wave32 handles 16 elements via V_WMMA_F32_16X16X4_F32:
//   A (16x4): lanes 0-15 hold K0/K1 (VGPR0/1), lanes 16-31 hold K2/K3; rows>=4 zero-padded.
//   B (4x16): lanes 0-15 hold K0/K1 of column (lane), lanes 16-31 hold K2/K3 of column (lane-16);
//             both halves load the same element's features (redundant loads, no shuffles needed).
//   D (16x16): VGPR j, lanes 0-15 = row j, col = lane  -> lane l owns element l's outputs.
// Layer1: K=10 padded to 12 -> 3 chained WMMAs.  Layer2: K=6 padded to 8 -> 2 chained WMMAs.
// Biases are folded into the accumulator's initial value (row j == VGPR j on lanes 0-15).
// Weight loads are branchless: row index clamped to a valid address, pad rows zeroed by cndmask.
__global__ void __launch_bounds__(128) tower_update(
    int count,
    const float* __restrict__ acc5,   // [i*5]: Ax (or Aty), [i*5+1..4]: A@h (or At@h)
    const float* __restrict__ bvec,   // b (dual) or c (primal)
    float sign,                       // dual: r = Ax - b (sign=-1); primal: d = Aty + c (sign=+1)
    float* __restrict__ st5,          // [i*5]: y/x, [i*5+1..4]: y_h/x_h  (updated in place)
    const float* __restrict__ sol,    // dual_solution / solution
    const float* __restrict__ W1, const float* __restrict__ B1,   // [4,10],[4]
    const float* __restrict__ W2, const float* __restrict__ B2,   // [4,6],[4]
    const float* __restrict__ WU, const float* __restrict__ BU,   // [6],[1]
    float* __restrict__ norm_acc) {
  const int tid  = threadIdx.x;
  const int lane = tid & 31;
  const int half = lane >> 4;       // which K-pair this lane supplies
  const int nl   = lane & 15;       // element slot within wave / matrix row for A
  const long gwave = ((long)blockIdx.x * blockDim.x + tid) >> 5;
  const int elem = (int)(gwave * 16) + nl;
  const int ei   = elem < count ? elem : count - 1;  // clamp: loads stay in range, EXEC stays full
  const long b5  = (long)ei * 5;

  // Per-element features (both wave halves load the same element's data)
  float av  = acc5[b5 + 0];
  float ah0 = acc5[b5 + 1], ah1 = acc5[b5 + 2], ah2 = acc5[b5 + 3], ah3 = acc5[b5 + 4];
  float sv  = st5[b5 + 0];
  float sh0 = st5[b5 + 1], sh1 = st5[b5 + 2], sh2 = st5[b5 + 3], sh3 = st5[b5 + 4];
  float rv  = av + sign * bvec[ei];

  const int  mrow  = nl;
  const bool rowok = (mrow < 4);
  const int  wrow  = rowok ? mrow : 0;   // clamped: unconditional loads from valid addresses

  // ---- layer 1: relu(W1 @ [sh0..3, ah0..3, sv, rv] + B1), K = 10 (pad 12) ----
  float w0x = W1[wrow * 10 + (half ? 2 : 0)];
  float w0y = W1[wrow * 10 + (half ? 3 : 1)];
  float w1x = W1[wrow * 10 + (half ? 6 : 4)];
  float w1y = W1[wrow * 10 + (half ? 7 : 5)];
  float w2x = W1[wrow * 10 + 8];
  float w2y = W1[wrow * 10 + 9];
  float a0x = rowok ? w0x : 0.f;
  float a0y = rowok ? w0y : 0.f;
  float a1x = rowok ? w1x : 0.f;
  float a1y = rowok ? w1y : 0.f;
  float a2x = (rowok && !half) ? w2x : 0.f;
  float a2y = (rowok && !half) ? w2y : 0.f;

  v2f A, Bm;
  v8f C = {};
  C[0] = B1[0]; C[1] = B1[1]; C[2] = B1[2]; C[3] = B1[3];   // bias pre-folded into rows 0-3
  A.x = a0x; A.y = a0y;
  Bm.x = half ? sh2 : sh0;  Bm.y = half ? sh3 : sh1;
  C = __builtin_amdgcn_wmma_f32_16x16x4_f32(false, A, false, Bm, (short)0, C, false, false);
  A.x = a1x; A.y = a1y;
  Bm.x = half ? ah2 : ah0;  Bm.y = half ? ah3 : ah1;
  C = __builtin_amdgcn_wmma_f32_16x16x4_f32(false, A, false, Bm, (short)0, C, false, false);
  A.x = a2x; A.y = a2y;
  Bm.x = half ? 0.f : sv;   Bm.y = half ? 0.f : rv;
  C = __builtin_amdgcn_wmma_f32_16x16x4_f32(false, A, false, Bm, (short)0, C, false, false);

  float t0 = fmaxf(C[0], 0.f);
  float t1 = fmaxf(C[1], 0.f);
  float t2 = fmaxf(C[2], 0.f);
  float t3 = fmaxf(C[3], 0.f);
  // broadcast element (lane&15)'s h1 so the upper half can supply K2/K3 of layer-2 B
  float e0 = __shfl(t0, nl, 32);
  float e1 = __shfl(t1, nl, 32);
  float e2 = __shfl(t2, nl, 32);
  float e3 = __shfl(t3, nl, 32);

  // ---- layer 2: relu(W2 @ [h1_0..3, sv, rv] + B2), K = 6 (pad 8) ----
  float v0x = W2[wrow * 6 + (half ? 2 : 0)];
  float v0y = W2[wrow * 6 + (half ? 3 : 1)];
  float v1x = W2[wrow * 6 + 4];
  float v1y = W2[wrow * 6 + 5];
  float c0x = rowok ? v0x : 0.f;
  float c0y = rowok ? v0y : 0.f;
  float c1x = (rowok && !half) ? v1x : 0.f;
  float c1y = (rowok && !half) ? v1y : 0.f;

  v8f C2 = {};
  C2[0] = B2[0]; C2[1] = B2[1]; C2[2] = B2[2]; C2[3] = B2[3];
  A.x = c0x; A.y = c0y;
  Bm.x = half ? e2 : e0;  Bm.y = half ? e3 : e1;
  C2 = __builtin_amdgcn_wmma_f32_16x16x4_f32(false, A, false, Bm, (short)0, C2, false, false);
  A.x = c1x; A.y = c1y;
  Bm.x = half ? 0.f : sv; Bm.y = half ? 0.f : rv;
  C2 = __builtin_amdgcn_wmma_f32_16x16x4_f32(false, A, false, Bm, (short)0, C2, false, false);

  float u0 = fmaxf(C2[0], 0.f);
  float u1 = fmaxf(C2[1], 0.f);
  float u2 = fmaxf(C2[2], 0.f);
  float u3 = fmaxf(C2[3], 0.f);

  // ---- update head: v += WU @ [u0..3, sv, rv] + BU ----
  float vnew = sv + BU[0] + WU[0] * u0 + WU[1] * u1 + WU[2] * u2 + WU[3] * u3
             + WU[4] * sv + WU[5] * rv;

  float local = 0.f;
  if (half == 0 && elem < count) {      // only the lower half owns valid lanes/elements
    st5[b5 + 0] = vnew;
    st5[b5 + 1] = u0; st5[b5 + 2] = u1; st5[b5 + 3] = u2; st5[b5 + 4] = u3;
    float d = vnew - sol[ei];
    local = d * d;
  }

  __shared__ float red[128];
  red[tid] = local;
  __syncthreads();
  for (int s = 64; s > 0; s >>= 1) {
    if (tid < s) red[tid] += red[tid + s];
    __syncthreads();
  }
  if (tid == 0) fatomic_add(norm_acc, red[0]);
}

__global__ void finalize_loss(const float* __restrict__ dp_acc, const float* __restrict__ dd_acc,
                              float* __restrict__ out, long off) {
  if (threadIdx.x != 0 || blockIdx.x != 0) return;
  float dps[NITER_FIXED], dds[NITER_FIXED];
  float sp = 0.f, sd = 0.f;
  for (int k = 0; k < NITER_FIXED; ++k) {
    dps[k] = sqrtf(dp_acc[k]);
    dds[k] = sqrtf(dd_acc[k]);
    sp += dps[k]; sd += dds[k];
  }
  float mp = 0.f, md = 0.f;
  for (int k = 1; k < NITER_FIXED; ++k) {
    mp += fmaxf(dps[k] - dps[k - 1], 0.f);
    md += fmaxf(dds[k] - dds[k - 1], 0.f);
  }
  float loss = sp / NITER_FIXED + mp / (NITER_FIXED - 1)
             + sd / NITER_FIXED + md / (NITER_FIXED - 1);
  out[off] = loss;
  for (int k = 0; k < NITER_FIXED; ++k) {
    out[off + 1 + k] = dps[k];
    out[off + 1 + NITER_FIXED + k] = dds[k];
  }
}

__global__ void copy_out(const float* __restrict__ x5, const float* __restrict__ y5,
                         float* __restrict__ out, int n, int m) {
  int i = blockIdx.x * blockDim.x + threadIdx.x;
  if (i < n)            out[i] = x5[(long)i * 5];
  else if (i < n + m)   out[i] = y5[(long)(i - n) * 5];
}

extern "C" void kernel_launch(void* const* d_in, const int* in_sizes, int n_in,
                              void* d_out, int out_size, void* d_ws, size_t ws_size,
                              hipStream_t stream) {
  const int*   rows          = (const int*)d_in[0];
  const int*   cols          = (const int*)d_in[1];
  const float* vals          = (const float*)d_in[2];
  const float* b             = (const float*)d_in[3];
  const float* c             = (const float*)d_in[4];
  const float* solution      = (const float*)d_in[5];
  const float* dual_solution = (const float*)d_in[6];
  const float* x0            = (const float*)d_in[7];
  const float* y0            = (const float*)d_in[8];
  const float* dw1 = (const float*)d_in[9];
  const float* db1 = (const float*)d_in[10];
  const float* dw2 = (const float*)d_in[11];
  const float* db2 = (const float*)d_in[12];
  const float* dwu = (const float*)d_in[13];
  const float* dbu = (const float*)d_in[14];
  const float* pw1 = (const float*)d_in[15];
  const float* pb1 = (const float*)d_in[16];
  const float* pw2 = (const float*)d_in[17];
  const float* pb2 = (const float*)d_in[18];
  const float* pwu = (const float*)d_in[19];
  const float* pbu = (const float*)d_in[20];

  const int nnz = in_sizes[2];
  const int m   = in_sizes[3];   // rows of A (b lives in R^m)
  const int n   = in_sizes[4];   // cols of A (c lives in R^n)

  char* ws = (char*)d_ws;
  size_t o = 0;
  float* x5   = (float*)(ws + o); o += (size_t)n * 5 * sizeof(float);  // [x, x_h(4)] interleaved
  float* y5   = (float*)(ws + o); o += (size_t)m * 5 * sizeof(float);  // [y, y_h(4)] interleaved
  float* Ax5  = (float*)(ws + o); o += (size_t)m * 5 * sizeof(float);  // A @ [x | x_h]
  float* Aty5 = (float*)(ws + o); o += (size_t)n * 5 * sizeof(float);  // A.T @ [y | y_h]
  o = (o + 255) & ~(size_t)255;
  float* dp_acc = (float*)(ws + o); o += NITER_FIXED * sizeof(float);
  float* dd_acc = (float*)(ws + o); o += NITER_FIXED * sizeof(float);  // contiguous with dp_acc

  const dim3 blk(256);
  init_state5<<<(n + 255) / 256, blk, 0, stream>>>(x5, x0, n);
  init_state5<<<(m + 255) / 256, blk, 0, stream>>>(y5, y0, m);
  zero_f32<<<1, 32, 0, stream>>>(dp_acc, 2 * NITER_FIXED);

  const long m5 = (long)m * 5, n5 = (long)n * 5;
  const int upd_blocks_m = (m + 63) / 64;   // 128 threads = 4 waves = 64 elements per block
  const int upd_blocks_n = (n + 63) / 64;
  const int spmv_blocks  = (nnz + 255) / 256;

  for (int t = 0; t < NITER_FIXED; ++t) {
    // dual: r = A@x - b, atxh = A@x_h ; update y, y_h ; accumulate ||y - y*||^2
    zero_f32<<<(int)((m5 + 255) / 256), blk, 0, stream>>>(Ax5, m5);
    spmv5<<<spmv_blocks, blk, 0, stream>>>(cols, rows, vals, x5, Ax5, nnz);
    tower_update<<<upd_blocks_m, 128, 0, stream>>>(m, Ax5, b, -1.f, y5, dual_solution,
                                                   dw1, db1, dw2, db2, dwu, dbu, dd_acc + t);
    // primal: d = c + A.T@y, atyh = A.T@y_h ; update x, x_h ; accumulate ||x - x*||^2
    zero_f32<<<(int)((n5 + 255) / 256), blk, 0, stream>>>(Aty5, n5);
    spmv5<<<spmv_blocks, blk, 0, stream>>>(rows, cols, vals, y5, Aty5, nnz);
    tower_update<<<upd_blocks_n, 128, 0, stream>>>(n, Aty5, c, +1.f, x5, solution,
                                                   pw1, pb1, pw2, pb2, pwu, pbu, dp_acc + t);
  }

  finalize_loss<<<1, 1, 0, stream>>>(dp_acc, dd_acc, (float*)d_out, (long)n + m);
  copy_out<<<(n + m + 255) / 256, blk, 0, stream>>>(x5, y5, (float*)d_out, n, m);
}